// BOTNet_65111704207447
// MI455X (gfx1250) — compile-verified
//
#include <hip/hip_runtime.h>
#include <hip/hip_bf16.h>
#include <math.h>

// ---------------------------------------------------------------------------
// MACE/BOTNet energy + forces, gfx1250 (wave32, WMMA).
// Live-dataflow-optimized: only the lm=0 channel of each interaction layer
// reaches the energy (sh[:,0]==1), so spherical harmonics are dead code and
// position-gradients flow only through edge lengths (radial features).
// Edge MLPs (ef[8] -> 64 -> 32) are evaluated with V_WMMA_F32_16X16X4_F32,
// 16 edges per wave, LDS-staged between GEMMs. Segment sums via f32 atomics
// (node arrays ~5MB -> resident in 192MB L2).
// ---------------------------------------------------------------------------

typedef __attribute__((ext_vector_type(2))) float v2f;
typedef __attribute__((ext_vector_type(8))) float v8f;

#define WPB 4              // waves per block (1 tile of 16 edges per wave)
#define TPB (WPB * 32)

__device__ __forceinline__ v8f wmma4(v2f a, v2f b, v8f c) {
  // D(16x16,f32) = A(16x4,f32) x B(4x16,f32) + C
  return __builtin_amdgcn_wmma_f32_16x16x4_f32(false, a, false, b, (short)0, c,
                                               false, false);
}

// hardware v_rcp_f32 (~1 ulp) instead of IEEE division
__device__ __forceinline__ float frcp_(float x) { return __builtin_amdgcn_rcpf(x); }
__device__ __forceinline__ float sigm_(float x) { return frcp_(1.0f + __expf(-x)); }
__device__ __forceinline__ float silu_(float x) { return x * sigm_(x); }
__device__ __forceinline__ float dsilu_(float x) {
  float s = sigm_(x);
  return s * (1.0f + x * (1.0f - s));
}

// Bessel(8) * polynomial cutoff (p=5) and d/d(len)
__device__ __forceinline__ void radial8(float len, float* ef, float* def) {
  const float invR = 0.2f, PI = 3.14159265358979323846f;
  float u = len * invR, invl = frcp_(len);
  const float pref = 0.63245553203f;  // sqrt(2/5)
  float fc = 0.0f, dfc = 0.0f;
  if (u < 1.0f) {
    float u2 = u * u, u4 = u2 * u2, u5 = u4 * u;
    fc = 1.0f - 21.0f * u5 + 35.0f * u5 * u - 15.0f * u5 * u2;
    float om = 1.0f - u;
    dfc = -105.0f * u4 * om * om * invR;  // d fc / d len
  }
#pragma unroll
  for (int n = 1; n <= 8; ++n) {
    float w = (float)n * PI * invR;
    float sv, cv;
    __sincosf(w * len, &sv, &cv);
    float b = pref * sv * invl;
    ef[n - 1] = b * fc;
    if (def) {
      float db = pref * (w * cv * len - sv) * invl * invl;
      def[n - 1] = db * fc + b * dfc;
    }
  }
}

struct FwdTile {
  float ef[16 * 9];
  float stage[16 * 65];  // 16x64 silu(z), padded stride (bank-conflict-free)
  int snd[16];
  int rcv[16];
};

struct BwdTile {
  float ef[16 * 9];      // ef, later reused for gef (16x8)
  float def[16 * 9];
  float uvec[16 * 3];
  float stage[16 * 65];  // silu(z) then gz*dsilu(z)
  float gt[16 * 33];     // 16x32 grad wrt t
  int snd[16];
  int rcv[16];
};

// ---------------------------------------------------------------------------
// Node kernels
// ---------------------------------------------------------------------------
__global__ void be_node_init(const float* __restrict__ attrs,
                             const float* __restrict__ Wemb,
                             const float* __restrict__ ae,
                             float* __restrict__ h, float* __restrict__ nE,
                             int N, int nelem) {
  int t = blockIdx.x * blockDim.x + threadIdx.x;
  if (t < N * 32) {
    int nd = t >> 5, c = t & 31;
    float acc = 0.0f;
    for (int k = 0; k < nelem; ++k) acc += attrs[nd * nelem + k] * Wemb[k * 32 + c];
    h[t] = acc;
    if (c == 0) {
      float e0 = 0.0f;
      for (int k = 0; k < nelem; ++k) e0 += attrs[nd * nelem + k] * ae[k];
      nE[nd] = e0;
    }
  }
}

__global__ void be_node_mid(const float* __restrict__ acc1,
                            const float* __restrict__ h,
                            const float* __restrict__ Wr1,
                            float* __restrict__ A1row0, float* __restrict__ nE,
                            int N) {
  int nd = blockIdx.x * blockDim.x + threadIdx.x;
  if (nd < N) {
    float e1 = 0.0f;
#pragma unroll
    for (int c = 0; c < 32; ++c) {
      float a = acc1[nd * 32 + c] * 0.0625f + h[nd * 32 + c];
      A1row0[nd * 32 + c] = a;
      e1 += a * Wr1[c];
    }
    nE[nd] += e1;
  }
}

__global__ void be_node_top(const float* __restrict__ acc2,
                            const float* __restrict__ A1row0,
                            const float* __restrict__ Wm1,
                            const float* __restrict__ Wm2,
                            const float* __restrict__ nE,
                            const int* __restrict__ batch,
                            float* __restrict__ g2,
                            float* __restrict__ out_total,
                            float* __restrict__ out_node, int N) {
  int nd = blockIdx.x * blockDim.x + threadIdx.x;
  if (nd < N) {
    float zk[16];
#pragma unroll
    for (int k = 0; k < 16; ++k) zk[k] = 0.0f;
    for (int c = 0; c < 32; ++c) {
      float a = acc2[nd * 32 + c] * 0.0625f + A1row0[nd * 32 + c];
#pragma unroll
      for (int k = 0; k < 16; ++k) zk[k] += a * Wm1[c * 16 + k];
    }
    float e2 = 0.0f;
    float gk[16];
#pragma unroll
    for (int k = 0; k < 16; ++k) {
      e2 += silu_(zk[k]) * Wm2[k];
      gk[k] = dsilu_(zk[k]) * Wm2[k];
    }
    float ne = nE[nd] + e2;
    out_node[nd] = ne;
    atomicAdd(&out_total[batch[nd]], ne);
    for (int c = 0; c < 32; ++c) {
      float g = 0.0f;
#pragma unroll
      for (int k = 0; k < 16; ++k) g += Wm1[c * 16 + k] * gk[k];
      g2[nd * 32 + c] = g;
    }
  }
}

__global__ void be_node_ga1(const float* __restrict__ g2,
                            const float* __restrict__ gh2,
                            const float* __restrict__ Wr1,
                            float* __restrict__ gA1, int N) {
  int t = blockIdx.x * blockDim.x + threadIdx.x;
  if (t < N * 32) gA1[t] = Wr1[t & 31] + g2[t] + gh2[t];
}

// ---------------------------------------------------------------------------
// Edge forward: acc[recv] += x[send] * (silu(ef@W1) @ W2[:,0:32])   (/16 later)
// ---------------------------------------------------------------------------
__global__ void __launch_bounds__(TPB)
be_edge_fwd(const float* __restrict__ pos, const float* __restrict__ shifts,
            const int* __restrict__ eidx, const float* __restrict__ xin,
            const float* __restrict__ W1, const float* __restrict__ W2,
            float* __restrict__ acc, int nEdges) {
  __shared__ float sW1[8 * 64];
  __shared__ float sW2[64 * 32];
  __shared__ FwdTile tiles[WPB];
  for (int i = threadIdx.x; i < 8 * 64; i += blockDim.x) sW1[i] = W1[i];
  for (int i = threadIdx.x; i < 64 * 32; i += blockDim.x)
    sW2[i] = W2[(i >> 5) * 96 + (i & 31)];

  FwdTile& T = tiles[threadIdx.x >> 5];
  int lane = threadIdx.x & 31;
  int tileBase = (blockIdx.x * WPB + (threadIdx.x >> 5)) * 16;

  if (lane < 16) {  // geometry: one edge per lane
    int e = tileBase + lane;
    if (e < nEdges) {
      int s = eidx[e], r = eidx[nEdges + e];
      float vx = pos[3 * r + 0] - pos[3 * s + 0] + shifts[3 * e + 0];
      float vy = pos[3 * r + 1] - pos[3 * s + 1] + shifts[3 * e + 1];
      float vz = pos[3 * r + 2] - pos[3 * s + 2] + shifts[3 * e + 2];
      float l = sqrtf(fmaxf(vx * vx + vy * vy + vz * vz, 1e-12f));
      float ef[8];
      radial8(l, ef, nullptr);
#pragma unroll
      for (int k = 0; k < 8; ++k) T.ef[lane * 9 + k] = ef[k];
      T.snd[lane] = s;
      T.rcv[lane] = r;
    } else {
#pragma unroll
      for (int k = 0; k < 8; ++k) T.ef[lane * 9 + k] = 0.0f;
      T.snd[lane] = -1;
      T.rcv[lane] = -1;
    }
  }
  __syncthreads();

  int mn = lane & 15;            // A-row / B-col within tile
  int ko = (lane >> 4) * 2;      // K sub-offset per A/B layout
  int mrow0 = 8 * (lane >> 4);   // D-layout row base

  // GEMM1: Z(16x64) = EF(16x8) @ W1(8x64)
  v8f z[4];
#pragma unroll
  for (int nb = 0; nb < 4; ++nb) z[nb] = (v8f){};
#pragma unroll
  for (int nb = 0; nb < 4; ++nb)
#pragma unroll
    for (int ks = 0; ks < 2; ++ks) {
      int k0 = ks * 4 + ko;
      v2f a, b;
      a.x = T.ef[mn * 9 + k0];
      a.y = T.ef[mn * 9 + k0 + 1];
      b.x = sW1[k0 * 64 + nb * 16 + mn];
      b.y = sW1[(k0 + 1) * 64 + nb * 16 + mn];
      z[nb] = wmma4(a, b, z[nb]);
    }
#pragma unroll
  for (int nb = 0; nb < 4; ++nb)
#pragma unroll
    for (int r = 0; r < 8; ++r)
      T.stage[(mrow0 + r) * 65 + nb * 16 + mn] = silu_(z[nb][r]);
  __syncthreads();

  // GEMM2: T1(16x32) = S(16x64) @ W2col(64x32)
  v8f t[2];
  t[0] = (v8f){};
  t[1] = (v8f){};
#pragma unroll
  for (int cb = 0; cb < 2; ++cb)
#pragma unroll
    for (int ks = 0; ks < 16; ++ks) {
      int k0 = ks * 4 + ko;
      v2f a, b;
      a.x = T.stage[mn * 65 + k0];
      a.y = T.stage[mn * 65 + k0 + 1];
      b.x = sW2[k0 * 32 + cb * 16 + mn];
      b.y = sW2[(k0 + 1) * 32 + cb * 16 + mn];
      t[cb] = wmma4(a, b, t[cb]);
    }

  // scatter: acc[recv] += x[send] * t   (D-layout: lane holds (mrow0+r, cb*16+mn))
#pragma unroll
  for (int cb = 0; cb < 2; ++cb)
#pragma unroll
    for (int r = 0; r < 8; ++r) {
      int mm = mrow0 + r;
      int s = T.snd[mm];
      if (s >= 0) {
        int rc = T.rcv[mm];
        int c = cb * 16 + mn;
        atomicAdd(&acc[rc * 32 + c], xin[s * 32 + c] * t[cb][r]);
      }
    }
}

// ---------------------------------------------------------------------------
// Edge backward for one layer:
//   gt_e = (1/16) * grad_r[recv] * x_s[send]        (grad wrt t_e)
//   gh  += (1/16) * grad_r[recv] * t_e  into gh_acc[send]   (layer2 only)
//   dE/dlen = ((gt @ W2colT) * dsilu(z) @ W1T) . def/dlen ; force atomics
// ---------------------------------------------------------------------------
__global__ void __launch_bounds__(TPB)
be_edge_bwd(const float* __restrict__ pos, const float* __restrict__ shifts,
            const int* __restrict__ eidx, const float* __restrict__ grad_r,
            const float* __restrict__ x_s, float* __restrict__ gh_acc,
            const float* __restrict__ W1, const float* __restrict__ W2,
            float* __restrict__ forces, int nEdges) {
  __shared__ float sW1[8 * 64];
  __shared__ float sW2[64 * 32];
  __shared__ BwdTile tiles[WPB];
  for (int i = threadIdx.x; i < 8 * 64; i += blockDim.x) sW1[i] = W1[i];
  for (int i = threadIdx.x; i < 64 * 32; i += blockDim.x)
    sW2[i] = W2[(i >> 5) * 96 + (i & 31)];

  BwdTile& T = tiles[threadIdx.x >> 5];
  int lane = threadIdx.x & 31;
  int tileBase = (blockIdx.x * WPB + (threadIdx.x >> 5)) * 16;

  if (lane < 16) {
    int e = tileBase + lane;
    if (e < nEdges) {
      int s = eidx[e], r = eidx[nEdges + e];
      float vx = pos[3 * r + 0] - pos[3 * s + 0] + shifts[3 * e + 0];
      float vy = pos[3 * r + 1] - pos[3 * s + 1] + shifts[3 * e + 1];
      float vz = pos[3 * r + 2] - pos[3 * s + 2] + shifts[3 * e + 2];
      float l = sqrtf(fmaxf(vx * vx + vy * vy + vz * vz, 1e-12f));
      float il = frcp_(l);
      T.uvec[lane * 3 + 0] = vx * il;
      T.uvec[lane * 3 + 1] = vy * il;
      T.uvec[lane * 3 + 2] = vz * il;
      float ef[8], df[8];
      radial8(l, ef, df);
#pragma unroll
      for (int k = 0; k < 8; ++k) {
        T.ef[lane * 9 + k] = ef[k];
        T.def[lane * 9 + k] = df[k];
      }
      T.snd[lane] = s;
      T.rcv[lane] = r;
    } else {
#pragma unroll
      for (int k = 0; k < 8; ++k) {
        T.ef[lane * 9 + k] = 0.0f;
        T.def[lane * 9 + k] = 0.0f;
      }
      T.uvec[lane * 3 + 0] = T.uvec[lane * 3 + 1] = T.uvec[lane * 3 + 2] = 0.0f;
      T.snd[lane] = -1;
      T.rcv[lane] = -1;
    }
  }
  __syncthreads();

  int mn = lane & 15;
  int ko = (lane >> 4) * 2;
  int mrow0 = 8 * (lane >> 4);

  // GEMM1 (recompute): Z = EF @ W1
  v8f z[4];
#pragma unroll
  for (int nb = 0; nb < 4; ++nb) z[nb] = (v8f){};
#pragma unroll
  for (int nb = 0; nb < 4; ++nb)
#pragma unroll
    for (int ks = 0; ks < 2; ++ks) {
      int k0 = ks * 4 + ko;
      v2f a, b;
      a.x = T.ef[mn * 9 + k0];
      a.y = T.ef[mn * 9 + k0 + 1];
      b.x = sW1[k0 * 64 + nb * 16 + mn];
      b.y = sW1[(k0 + 1) * 64 + nb * 16 + mn];
      z[nb] = wmma4(a, b, z[nb]);
    }
#pragma unroll
  for (int nb = 0; nb < 4; ++nb)
#pragma unroll
    for (int r = 0; r < 8; ++r)
      T.stage[(mrow0 + r) * 65 + nb * 16 + mn] = silu_(z[nb][r]);
  __syncthreads();

  // GEMM2 (recompute): t = S @ W2col
  v8f t[2];
  t[0] = (v8f){};
  t[1] = (v8f){};
#pragma unroll
  for (int cb = 0; cb < 2; ++cb)
#pragma unroll
    for (int ks = 0; ks < 16; ++ks) {
      int k0 = ks * 4 + ko;
      v2f a, b;
      a.x = T.stage[mn * 65 + k0];
      a.y = T.stage[mn * 65 + k0 + 1];
      b.x = sW2[k0 * 32 + cb * 16 + mn];
      b.y = sW2[(k0 + 1) * 32 + cb * 16 + mn];
      t[cb] = wmma4(a, b, t[cb]);
    }

  // gt tile + gh scatter
#pragma unroll
  for (int cb = 0; cb < 2; ++cb)
#pragma unroll
    for (int r = 0; r < 8; ++r) {
      int mm = mrow0 + r;
      int s = T.snd[mm];
      float gtv = 0.0f;
      if (s >= 0) {
        int rc = T.rcv[mm];
        int c = cb * 16 + mn;
        float gr = grad_r[rc * 32 + c];
        if (gh_acc) atomicAdd(&gh_acc[s * 32 + c], 0.0625f * gr * t[cb][r]);
        gtv = 0.0625f * gr * x_s[s * 32 + c];
      }
      T.gt[mm * 33 + cb * 16 + mn] = gtv;
    }
  __syncthreads();

  // GEMM3: gz_pre(16x64) = GT(16x32) @ W2colT(32x64);  gz = gz_pre * dsilu(z)
  v8f gz[4];
#pragma unroll
  for (int nb = 0; nb < 4; ++nb) gz[nb] = (v8f){};
#pragma unroll
  for (int nb = 0; nb < 4; ++nb)
#pragma unroll
    for (int ks = 0; ks < 8; ++ks) {
      int k0 = ks * 4 + ko;
      v2f a, b;
      a.x = T.gt[mn * 33 + k0];
      a.y = T.gt[mn * 33 + k0 + 1];
      b.x = sW2[(nb * 16 + mn) * 32 + k0];      // W2colT[k][j] = W2col[j][k]
      b.y = sW2[(nb * 16 + mn) * 32 + k0 + 1];
      gz[nb] = wmma4(a, b, gz[nb]);
    }
#pragma unroll
  for (int nb = 0; nb < 4; ++nb)
#pragma unroll
    for (int r = 0; r < 8; ++r)
      T.stage[(mrow0 + r) * 65 + nb * 16 + mn] = gz[nb][r] * dsilu_(z[nb][r]);
  __syncthreads();

  // GEMM4: gef(16x16; cols 0..7 valid) = GZ(16x64) @ W1T(64x16, zero-padded)
  v8f ge = (v8f){};
#pragma unroll
  for (int ks = 0; ks < 16; ++ks) {
    int k0 = ks * 4 + ko;
    v2f a, b;
    a.x = T.stage[mn * 65 + k0];
    a.y = T.stage[mn * 65 + k0 + 1];
    b.x = (mn < 8) ? sW1[mn * 64 + k0] : 0.0f;       // W1T[k][n] = W1[n][k]
    b.y = (mn < 8) ? sW1[mn * 64 + k0 + 1] : 0.0f;
    ge = wmma4(a, b, ge);
  }
  // stage gef (reuse T.ef)
#pragma unroll
  for (int r = 0; r < 8; ++r)
    if (mn < 8) T.ef[(mrow0 + r) * 9 + mn] = ge[r];
  __syncthreads();

  // dE/dlen = gef . def/dlen ; force atomics (F = -dE/dpos)
  if (lane < 16) {
    int mm = lane;
    int s = T.snd[mm];
    if (s >= 0) {
      int rc = T.rcv[mm];
      float g = 0.0f;
#pragma unroll
      for (int k = 0; k < 8; ++k) g += T.ef[mm * 9 + k] * T.def[mm * 9 + k];
#pragma unroll
      for (int d = 0; d < 3; ++d) {
        float fv = g * T.uvec[mm * 3 + d];
        atomicAdd(&forces[rc * 3 + d], -fv);
        atomicAdd(&forces[s * 3 + d], fv);
      }
    }
  }
}

// ---------------------------------------------------------------------------
extern "C" void kernel_launch(void* const* d_in, const int* in_sizes, int n_in,
                              void* d_out, int out_size, void* d_ws,
                              size_t ws_size, hipStream_t stream) {
  const float* pos = (const float*)d_in[0];
  const float* attrs = (const float*)d_in[1];
  const float* shifts = (const float*)d_in[2];
  const float* ae = (const float*)d_in[3];
  const float* Wemb = (const float*)d_in[4];
  const float* W1a = (const float*)d_in[5];
  const float* W2a = (const float*)d_in[6];
  const float* Wr1 = (const float*)d_in[7];
  const float* W1b = (const float*)d_in[8];
  const float* W2b = (const float*)d_in[9];
  const float* Wm1 = (const float*)d_in[10];
  const float* Wm2 = (const float*)d_in[11];
  const int* eidx = (const int*)d_in[12];
  const int* batch = (const int*)d_in[13];

  const int N = in_sizes[0] / 3;
  const int nelem = in_sizes[1] / N;
  const int E = in_sizes[12] / 2;
  const int ngraph = out_size - 4 * N;

  float* ws = (float*)d_ws;
  size_t o = 0;
  float* h = ws + o;      o += (size_t)N * 32;
  float* A1row0 = ws + o; o += (size_t)N * 32;
  float* g2 = ws + o;     o += (size_t)N * 32;
  float* gA1 = ws + o;    o += (size_t)N * 32;
  float* nE = ws + o;     o += (size_t)N;
  float* acc1 = ws + o;   o += (size_t)N * 32;
  float* acc2 = ws + o;   o += (size_t)N * 32;
  float* gh2 = ws + o;    o += (size_t)N * 32;

  float* out_total = (float*)d_out;
  float* out_node = out_total + ngraph;
  float* out_forces = out_node + N;

  // zero outputs + edge-scatter accumulators (acc1,acc2,gh2 are contiguous)
  hipMemsetAsync(d_out, 0, (size_t)out_size * sizeof(float), stream);
  hipMemsetAsync(acc1, 0, (size_t)N * 32 * 3 * sizeof(float), stream);

  const int nThread = 256;
  const int edgeBlocks = (E + 16 * WPB - 1) / (16 * WPB);

  be_node_init<<<(N * 32 + nThread - 1) / nThread, nThread, 0, stream>>>(
      attrs, Wemb, ae, h, nE, N, nelem);

  be_edge_fwd<<<edgeBlocks, TPB, 0, stream>>>(pos, shifts, eidx, h, W1a, W2a,
                                              acc1, E);
  be_node_mid<<<(N + nThread - 1) / nThread, nThread, 0, stream>>>(
      acc1, h, Wr1, A1row0, nE, N);

  be_edge_fwd<<<edgeBlocks, TPB, 0, stream>>>(pos, shifts, eidx, A1row0, W1b,
                                              W2b, acc2, E);
  be_node_top<<<(N + nThread - 1) / nThread, nThread, 0, stream>>>(
      acc2, A1row0, Wm1, Wm2, nE, batch, g2, out_total, out_node, N);

  // layer-2 backward (also accumulates gh2 into senders)
  be_edge_bwd<<<edgeBlocks, TPB, 0, stream>>>(pos, shifts, eidx, g2, A1row0,
                                              gh2, W1b, W2b, out_forces, E);
  be_node_ga1<<<(N * 32 + nThread - 1) / nThread, nThread, 0, stream>>>(
      g2, gh2, Wr1, gA1, N);

  // layer-1 backward (h is position-independent -> no gh accumulation)
  be_edge_bwd<<<edgeBlocks, TPB, 0, stream>>>(pos, shifts, eidx, gA1, h,
                                              nullptr, W1a, W2a, out_forces, E);
}